// SGC_33346126086445
// MI455X (gfx1250) — compile-verified
//
#include <hip/hip_runtime.h>
#include <hip/hip_bf16.h>

// ---------------------------------------------------------------------------
// SGC: h = Â^2 x ; out = h @ W^T + b, with Â = D^-1/2 (A + I) D^-1/2
// N=100000, E=600000, D=128, K=2.
// Propagation is L2-resident scatter/gather (h = 51.2MB < 192MB L2).
// Final projection done with V_WMMA_F32_16X16X4_F32 (16-node tiles).
// ---------------------------------------------------------------------------

#define DIM 128

typedef __attribute__((ext_vector_type(2))) float v2f;
typedef __attribute__((ext_vector_type(8))) float v8f;

// --- degree pipeline --------------------------------------------------------

// dinv[i] = 1.0 (self-loop contribution to degree)
__global__ void sgc_deg_init(float* __restrict__ dinv, int N) {
    int i = blockIdx.x * blockDim.x + threadIdx.x;
    if (i < N) dinv[i] = 1.0f;
}

// dinv[col[e]] += 1 for each real edge
__global__ void sgc_deg_count(const int* __restrict__ col, float* __restrict__ dinv, int E) {
    int e = blockIdx.x * blockDim.x + threadIdx.x;
    if (e < E) atomicAdd(&dinv[col[e]], 1.0f);
}

// dinv[i] = rsqrt(deg[i])  (deg >= 1 always, self loops guarantee it)
__global__ void sgc_deg_rsqrt(float* __restrict__ dinv, int N) {
    int i = blockIdx.x * blockDim.x + threadIdx.x;
    if (i < N) dinv[i] = rsqrtf(dinv[i]);
}

// --- propagation ------------------------------------------------------------

// h_out[i,:] = dinv[i]^2 * h_in[i,:]   (self-loop term; also initializes h_out)
// one float4 per thread, 32 threads per node row
__global__ void sgc_self_init(const float* __restrict__ hin,
                              const float* __restrict__ dinv,
                              float* __restrict__ hout, int N) {
    int t = blockIdx.x * blockDim.x + threadIdx.x;
    int i = t >> 5;
    if (i >= N) return;
    int q = t & 31;
    float s = dinv[i] * dinv[i];
    float4 v = ((const float4*)(hin + (size_t)i * DIM))[q];
    v.x *= s; v.y *= s; v.z *= s; v.w *= s;
    ((float4*)(hout + (size_t)i * DIM))[q] = v;
}

// h_out[col[e],:] += dinv[row]*dinv[col] * h_in[row[e],:]
// one wave (32 lanes) per edge; each lane: B128 load + 4 f32 atomics.
__global__ void sgc_edge_scatter(const int* __restrict__ row,
                                 const int* __restrict__ col,
                                 const float* __restrict__ dinv,
                                 const float* __restrict__ hin,
                                 float* __restrict__ hout, int E) {
    int t = blockIdx.x * blockDim.x + threadIdx.x;
    int e = t >> 5;
    if (e >= E) return;
    int q = t & 31;
    int r = row[e];
    int c = col[e];
    float nrm = dinv[r] * dinv[c];
    float4 v = ((const float4*)(hin + (size_t)r * DIM))[q];
    float* dst = hout + (size_t)c * DIM + q * 4;
    atomicAdd(dst + 0, v.x * nrm);
    atomicAdd(dst + 1, v.y * nrm);
    atomicAdd(dst + 2, v.z * nrm);
    atomicAdd(dst + 3, v.w * nrm);
}

// --- final projection via WMMA ---------------------------------------------
// out[i] = dot(h[i,:], W) + b using V_WMMA_F32_16X16X4_F32.
// One wave handles a 16-node tile; K=128 consumed in chunks of 4.
// A (16x4 f32): lanes 0-15 hold K=0 (vgpr0) / K=1 (vgpr1); lanes 16-31 K=2/K=3.
// B (4x16): W[k] broadcast along all N columns -> every output column equals
// the desired dot product (robust to N-lane mapping details).
// D (16x16 f32): vgpr j: lanes 0-15 -> M=j, lanes 16-31 -> M=j+8.
__global__ void sgc_gemv_wmma(const float* __restrict__ h,
                              const float* __restrict__ W,
                              const float* __restrict__ bias,
                              float* __restrict__ out, int nTiles) {
    const int lane = threadIdx.x & 31;
    const int wave = (int)((blockIdx.x * blockDim.x + threadIdx.x) >> 5);
    if (wave >= nTiles) return;            // uniform per wave -> EXEC all-1s below

    const int node0 = wave * 16;
    const int m     = lane & 15;           // row within tile
    const int koff  = (lane >> 4) << 1;    // 0 for lanes 0-15, 2 for lanes 16-31
    const float* hrow = h + (size_t)(node0 + m) * DIM;

    v8f c = {};
    #pragma unroll
    for (int k0 = 0; k0 < DIM; k0 += 4) {
        v2f a, bm;
        a.x  = hrow[k0 + koff];
        a.y  = hrow[k0 + koff + 1];
        bm.x = W[k0 + koff];
        bm.y = W[k0 + koff + 1];
        // (neg_a, A, neg_b, B, c_mod, C, reuse_a, reuse_b)
        c = __builtin_amdgcn_wmma_f32_16x16x4_f32(
                false, a, false, bm, (short)0, c, false, false);
    }

    float bb = bias[0];
    if (lane == 0) {
        #pragma unroll
        for (int j = 0; j < 8; ++j) out[node0 + j] = c[j] + bb;
    } else if (lane == 16) {
        #pragma unroll
        for (int j = 0; j < 8; ++j) out[node0 + 8 + j] = c[j] + bb;
    }
}

// scalar tail for N not divisible by 16 (not launched for N=100000)
__global__ void sgc_gemv_tail(const float* __restrict__ h,
                              const float* __restrict__ W,
                              const float* __restrict__ bias,
                              float* __restrict__ out, int start, int N) {
    int i = start + blockIdx.x * blockDim.x + threadIdx.x;
    if (i >= N) return;
    float acc = 0.0f;
    for (int k = 0; k < DIM; ++k) acc += h[(size_t)i * DIM + k] * W[k];
    out[i] = acc + bias[0];
}

// ---------------------------------------------------------------------------

extern "C" void kernel_launch(void* const* d_in, const int* in_sizes, int n_in,
                              void* d_out, int out_size, void* d_ws, size_t ws_size,
                              hipStream_t stream) {
    const float* x  = (const float*)d_in[0];           // [N, 128]
    const int*   ei = (const int*)d_in[1];             // [2, E] flat
    const float* W  = (const float*)d_in[2];           // [1, 128]
    const float* b  = (const float*)d_in[3];           // [1]
    float*       out = (float*)d_out;                  // [N]

    const int N = in_sizes[0] / DIM;
    const int E = in_sizes[1] / 2;
    const int* row = ei;        // edge_index[0] = source
    const int* col = ei + E;    // edge_index[1] = destination

    // workspace layout: hA | hB | dinv
    float* hA   = (float*)d_ws;
    float* hB   = hA + (size_t)N * DIM;
    float* dinv = hB + (size_t)N * DIM;

    const int TB = 256;
    dim3 blk(TB);

    // 1) degree -> dinv
    sgc_deg_init <<<dim3((N + TB - 1) / TB), blk, 0, stream>>>(dinv, N);
    sgc_deg_count<<<dim3((E + TB - 1) / TB), blk, 0, stream>>>(col, dinv, E);
    sgc_deg_rsqrt<<<dim3((N + TB - 1) / TB), blk, 0, stream>>>(dinv, N);

    // threads for per-node-row (32/node) and per-edge (32/edge) kernels
    const long long nodeThreads = (long long)N * 32;
    const long long edgeThreads = (long long)E * 32;
    dim3 gNode((unsigned)((nodeThreads + TB - 1) / TB));
    dim3 gEdge((unsigned)((edgeThreads + TB - 1) / TB));

    // 2) propagation step 1: x -> hA
    sgc_self_init   <<<gNode, blk, 0, stream>>>(x, dinv, hA, N);
    sgc_edge_scatter<<<gEdge, blk, 0, stream>>>(row, col, dinv, x, hA, E);

    // 3) propagation step 2: hA -> hB
    sgc_self_init   <<<gNode, blk, 0, stream>>>(hA, dinv, hB, N);
    sgc_edge_scatter<<<gEdge, blk, 0, stream>>>(row, col, dinv, hA, hB, E);

    // 4) projection: out = hB @ W^T + b   (WMMA, 16 nodes per wave)
    const int nTiles = N / 16;
    if (nTiles > 0) {
        const int wavesPerBlock = TB / 32;
        dim3 gGemv((unsigned)((nTiles + wavesPerBlock - 1) / wavesPerBlock));
        sgc_gemv_wmma<<<gGemv, blk, 0, stream>>>(hB, W, b, out, nTiles);
    }
    const int rem = N - nTiles * 16;
    if (rem > 0) {
        sgc_gemv_tail<<<dim3(1), dim3(64), 0, stream>>>(hB, W, b, out, nTiles * 16, N);
    }
}